// HaterogenicGraphNodeEncoder_84817014162126
// MI455X (gfx1250) — compile-verified
//
#include <hip/hip_runtime.h>
#include <hip/hip_bf16.h>

// ---------------------------------------------------------------------------
// GIN 2-layer encoder for MI455X (gfx1250, wave32).
// GEMMs use V_WMMA_F32_16X16X4_F32 (full fp32 precision; compute is ~13 GFLOP,
// i.e. negligible vs the ~2-3 GB of edge scatter/gather traffic at 23.3 TB/s,
// so f32 WMMA is both exact and free). Edge aggregation uses f32 global
// atomics. BN is folded into a per-column scale/shift epilogue.
// ---------------------------------------------------------------------------

typedef __attribute__((ext_vector_type(2))) float v2f;
typedef __attribute__((ext_vector_type(8))) float v8f;

#define BN_EPS 1e-5f

// ---- fold bias + batchnorm into per-column scale s and shift t -------------
__global__ void bn_fold_kernel(const float* __restrict__ b,
                               const float* __restrict__ gamma,
                               const float* __restrict__ beta,
                               const float* __restrict__ mean,
                               const float* __restrict__ var,
                               float* __restrict__ s, float* __restrict__ t,
                               int n) {
  int i = blockIdx.x * blockDim.x + threadIdx.x;
  if (i < n) {
    float sc = gamma[i] * rsqrtf(var[i] + BN_EPS);
    s[i] = sc;
    t[i] = (b[i] - mean[i]) * sc + beta[i];
  }
}

// ---- x0 = emb_table[node_ids], F = 64 (float4 granularity) -----------------
__global__ void gather_kernel(const float* __restrict__ emb,
                              const int* __restrict__ ids,
                              float* __restrict__ out, int n_nodes) {
  int t = blockIdx.x * blockDim.x + threadIdx.x;        // one float4 each
  int total = n_nodes * 16;                             // 64 floats / node
  if (t >= total) return;
  int i = t >> 4;
  int c = (t & 15) << 2;
  const float4 v = *(const float4*)(emb + (size_t)ids[i] * 64 + c);
  *(float4*)(out + (size_t)i * 64 + c) = v;
}

// ---- zero fill (float4) ----------------------------------------------------
__global__ void zero_kernel(float* __restrict__ p, int n_f4) {
  int t = blockIdx.x * blockDim.x + threadIdx.x;
  if (t < n_f4) *(float4*)(p + (size_t)t * 4) = make_float4(0.f, 0.f, 0.f, 0.f);
}

// ---- agg[dst] += x[src], F features (float4 load, 4 f32 atomics) ----------
template <int F>
__global__ void scatter_add_kernel(const float* __restrict__ x,
                                   const int* __restrict__ src,
                                   const int* __restrict__ dst,
                                   float* __restrict__ agg, int n_edges) {
  constexpr int CH = F >> 2;
  int t = blockIdx.x * blockDim.x + threadIdx.x;
  int total = n_edges * CH;
  if (t >= total) return;
  int e = t / CH;
  int c = (t - e * CH) << 2;
  int s = src[e];
  int d = dst[e];
  const float4 v = *(const float4*)(x + (size_t)s * F + c);
  float* p = agg + (size_t)d * F + c;
  atomicAdd(p + 0, v.x);
  atomicAdd(p + 1, v.y);
  atomicAdd(p + 2, v.z);
  atomicAdd(p + 3, v.w);
}

// ---- WMMA fp32 GEMM: out[N x M] = epilogue((A (+Aadd)) [N x K] @ B [K x M])
// One block = 16-row strip (staged in LDS, fused x+agg add); one wave = one
// 16x16 output tile; K-loop in steps of 4 via v_wmma_f32_16x16x4_f32.
// Epilogue: val = h*scale[col] + bias[col], optional ReLU.
// Requires N % 16 == 0 and blockDim.x == 32*(M/16) (EXEC all-ones for WMMA).
template <int K, int M>
__global__ void __launch_bounds__(256) gemm_wmma_kernel(
    const float* __restrict__ A, const float* __restrict__ Aadd,
    const float* __restrict__ B, const float* __restrict__ scale,
    const float* __restrict__ bias, float* __restrict__ out, int relu) {
  __shared__ float As[16 * K];
  const int tid = threadIdx.x;
  const int lane = tid & 31;
  const int wave = tid >> 5;
  const size_t rowBase = (size_t)blockIdx.x * 16;

  // Stage A strip (+ fused GIN self+aggregate add) into LDS.
  for (int idx = tid; idx < 16 * K; idx += blockDim.x) {
    int r = idx / K;
    int k = idx - r * K;
    size_t g = (rowBase + r) * (size_t)K + k;
    float v = A[g];
    if (Aadd) v += Aadd[g];
    As[idx] = v;
  }
  __syncthreads();

  const int colBase = wave * 16;
  const int half = lane >> 4;   // ISA: lanes 16-31 carry K+2 (A) / rows K+2 (B)
  const int l15 = lane & 15;

  v8f c = {};
#pragma unroll
  for (int k = 0; k < K; k += 4) {
    const int ka = k + half * 2;
    v2f a, b;
    a.x = As[l15 * K + ka];           // A[m=l15, ka]
    a.y = As[l15 * K + ka + 1];       // A[m=l15, ka+1]
    const float* Bp = B + (size_t)ka * M + colBase + l15;
    b.x = Bp[0];                      // B[ka,   n=l15]
    b.y = Bp[M];                      // B[ka+1, n=l15]
    c = __builtin_amdgcn_wmma_f32_16x16x4_f32(false, a, false, b,
                                              (short)0, c, false, false);
  }

  // Epilogue per ISA C/D layout: VGPR v -> M=v (lanes 0-15) / M=v+8 (16-31).
  const int col = colBase + l15;
  const float s = scale ? scale[col] : 1.0f;
  const float t = bias ? bias[col] : 0.0f;
#pragma unroll
  for (int v = 0; v < 8; ++v) {
    float val = c[v] * s + t;
    if (relu) val = fmaxf(val, 0.0f);
    size_t row = rowBase + v + half * 8;
    out[row * (size_t)M + col] = val;
  }
}

// ---------------------------------------------------------------------------
extern "C" void kernel_launch(void* const* d_in, const int* in_sizes, int n_in,
                              void* d_out, int out_size, void* d_ws,
                              size_t ws_size, hipStream_t stream) {
  const int N = in_sizes[0];          // 100000 (multiple of 16)
  const int E = in_sizes[1] / 2;      // 1600000

  const int* node_ids = (const int*)d_in[0];
  const int* src = (const int*)d_in[1];
  const int* dst = src + E;
  const float* emb = (const float*)d_in[2];
  const float* c1_W1 = (const float*)d_in[3];
  const float* c1_b1 = (const float*)d_in[4];
  const float* c1_gamma = (const float*)d_in[5];
  const float* c1_beta = (const float*)d_in[6];
  const float* c1_mean = (const float*)d_in[7];
  const float* c1_var = (const float*)d_in[8];
  const float* c1_W2 = (const float*)d_in[9];
  const float* c1_b2 = (const float*)d_in[10];
  const float* c2_W1 = (const float*)d_in[11];
  const float* c2_b1 = (const float*)d_in[12];
  const float* c2_gamma = (const float*)d_in[13];
  const float* c2_beta = (const float*)d_in[14];
  const float* c2_mean = (const float*)d_in[15];
  const float* c2_var = (const float*)d_in[16];
  const float* c2_W2 = (const float*)d_in[17];
  const float* c2_b2 = (const float*)d_in[18];
  const float* out_W = (const float*)d_in[19];
  const float* out_b = (const float*)d_in[20];
  float* out = (float*)d_out;

  // Workspace: 3 ping-pong buffers of N x 128 f32, plus folded BN params.
  float* ws = (float*)d_ws;
  const size_t per = (size_t)N * 128;
  float* R0 = ws;
  float* R1 = ws + per;
  float* R2 = ws + 2 * per;
  float* s1 = ws + 3 * per;
  float* t1 = s1 + 128;
  float* s2 = t1 + 128;
  float* t2 = s2 + 128;

  const int TPB = 256;
  const dim3 gemmGrid(N / 16);

  // Fold BN params for both conv MLPs.
  bn_fold_kernel<<<1, 128, 0, stream>>>(c1_b1, c1_gamma, c1_beta, c1_mean,
                                        c1_var, s1, t1, 128);
  bn_fold_kernel<<<1, 128, 0, stream>>>(c2_b1, c2_gamma, c2_beta, c2_mean,
                                        c2_var, s2, t2, 128);

  // x0 = emb[node_ids]  -> R0 (N x 64)
  {
    int total = N * 16;
    gather_kernel<<<(total + TPB - 1) / TPB, TPB, 0, stream>>>(emb, node_ids,
                                                               R0, N);
  }

  // ---- Layer 1 ----
  {
    int f4 = (N * 64) / 4;
    zero_kernel<<<(f4 + TPB - 1) / TPB, TPB, 0, stream>>>(R1, f4);
    int total = E * 16;  // 64/4 chunks per edge
    scatter_add_kernel<64><<<(total + TPB - 1) / TPB, TPB, 0, stream>>>(
        R0, src, dst, R1, E);
  }
  // h = relu(BN((x0+agg) @ c1_W1 + b1))      -> R2 (N x 128)
  gemm_wmma_kernel<64, 128><<<gemmGrid, 256, 0, stream>>>(R0, R1, c1_W1, s1,
                                                          t1, R2, 1);
  // x1 = relu(h @ c1_W2 + b2)                -> R0 (N x 128)
  gemm_wmma_kernel<128, 128><<<gemmGrid, 256, 0, stream>>>(
      R2, nullptr, c1_W2, nullptr, c1_b2, R0, 1);

  // ---- Layer 2 ----
  {
    int f4 = (N * 128) / 4;
    zero_kernel<<<(f4 + TPB - 1) / TPB, TPB, 0, stream>>>(R1, f4);
    int total = E * 32;  // 128/4 chunks per edge
    scatter_add_kernel<128><<<(total + TPB - 1) / TPB, TPB, 0, stream>>>(
        R0, src, dst, R1, E);
  }
  // h2 = relu(BN((x1+agg2) @ c2_W1 + b1))    -> R2 (N x 128)
  gemm_wmma_kernel<128, 128><<<gemmGrid, 256, 0, stream>>>(R0, R1, c2_W1, s2,
                                                           t2, R2, 1);
  // x2 = relu(h2 @ c2_W2 + b2)               -> R1 (N x 128)
  gemm_wmma_kernel<128, 128><<<gemmGrid, 256, 0, stream>>>(
      R2, nullptr, c2_W2, nullptr, c2_b2, R1, 1);

  // ---- Output head: out = x2 @ out_W + out_b  (N x 64, no relu) ----
  gemm_wmma_kernel<128, 64><<<gemmGrid, 128, 0, stream>>>(
      R1, nullptr, out_W, nullptr, out_b, out, 0);
}